// MessagePassing_87144886436049
// MI455X (gfx1250) — compile-verified
//
#include <hip/hip_runtime.h>
#include <hip/hip_bf16.h>

// Message passing / segment-sum scatter-add:
//   out[dst[e], :] += x[src[e], :]   for e in [0, N_EDGES)
// D_FEAT = 64 f32 per row. Bandwidth/atomic-bound (AI ~ 0.25 FLOP/B):
// x (25.6MB) and out (25.6MB) are L2-resident (192MB L2), so the kernel is
// limited by L2 gather + atomic throughput. WMMA is inapplicable (no dense
// GEMM structure); the gfx1250 paths that matter are non-returning
// global_atomic_add_f32, b128 coalesced gathers, NT hints on the streamed
// index arrays, and global_prefetch_b8.

#define DFEAT 64
#define CHUNKS 16  // 16 x float4 per 64-float row

__global__ __launch_bounds__(256) void MP_zero_kernel(float4* __restrict__ out, int n4) {
    int gid = blockIdx.x * 256 + threadIdx.x;
    if (gid < n4) {
        out[gid] = make_float4(0.f, 0.f, 0.f, 0.f);
    }
}

__global__ __launch_bounds__(256) void MP_scatter_kernel(
    const float* __restrict__ x,
    const int*   __restrict__ dst_idx,   // edge_index[0, :]
    const int*   __restrict__ src_idx,   // edge_index[1, :]
    float*       __restrict__ out,
    int n_edges)
{
    int gid = blockIdx.x * 256 + threadIdx.x;
    int e = gid >> 4;          // 16 lanes cooperate on one edge
    if (e >= n_edges) return;
    int c = gid & 15;          // float4 chunk within the 64-float row

    // Index stream is read exactly once -> non-temporal so it doesn't evict
    // the L2-resident x/out working set.
    int dst = __builtin_nontemporal_load(dst_idx + e);
    int src = __builtin_nontemporal_load(src_idx + e);

    // Prefetch the index stream ahead (one lane per edge-group keeps the
    // prefetch rate at ~1 per 16 lanes per array).
    if (c == 0) {
        __builtin_prefetch(dst_idx + e + 2048, 0, 1);
        __builtin_prefetch(src_idx + e + 2048, 0, 1);
    }

    // Coalesced 128-bit gather: 16 lanes x 16B = the full 256B row of x[src].
    const float4 v = *reinterpret_cast<const float4*>(
        x + (size_t)src * DFEAT + (size_t)c * 4);

    float* o = out + (size_t)dst * DFEAT + (size_t)c * 4;

    // Non-returning hardware f32 atomics (global_atomic_add_f32, STOREcnt).
    unsafeAtomicAdd(o + 0, v.x);
    unsafeAtomicAdd(o + 1, v.y);
    unsafeAtomicAdd(o + 2, v.z);
    unsafeAtomicAdd(o + 3, v.w);
}

extern "C" void kernel_launch(void* const* d_in, const int* in_sizes, int n_in,
                              void* d_out, int out_size, void* d_ws, size_t ws_size,
                              hipStream_t stream) {
    const float* x  = (const float*)d_in[0];
    const int*   ei = (const int*)d_in[1];      // [2, N_EDGES] flat
    const int n_edges = in_sizes[1] / 2;
    const int* dst_idx = ei;                    // row 0: destinations (segment ids)
    const int* src_idx = ei + n_edges;          // row 1: sources (gather ids)
    float* out = (float*)d_out;

    // 1) Zero the accumulator (harness poisons d_out).
    int n4 = out_size / 4;
    int zblocks = (n4 + 255) / 256;
    MP_zero_kernel<<<zblocks, 256, 0, stream>>>((float4*)out, n4);

    // 2) Scatter-add: 16 lanes per edge.
    long long total = (long long)n_edges * CHUNKS;   // 20M lanes
    int sblocks = (int)((total + 255) / 256);
    MP_scatter_kernel<<<sblocks, 256, 0, stream>>>(x, dst_idx, src_idx, out, n_edges);
}